// AttentionDecoder_40913858462225
// MI455X (gfx1250) — compile-verified
//
#include <hip/hip_runtime.h>
#include <hip/hip_bf16.h>

// ---------------------------------------------------------------------------
// AttentionDecoder step, MI455X (gfx1250, wave32).
// Roofline: ~603MB fp32 weight streaming @ 23.3TB/s ==> ~26us floor; all
// GEMVs are HBM-bound. GEMVs use V_WMMA_F32_16X16X4_F32: one wave = 16
// output rows; per K-step of 4, A = 16x4 weight tile (global b64/lane),
// B = input chunk broadcast across all 16 columns from LDS (ds b64
// broadcast). 8 waves/block split K; each wave runs chunks of 64 K
// (16 b64 loads clause-batched, 4KB in flight) feeding two alternating
// WMMA accumulators to break the D->C dependency chain.
// ---------------------------------------------------------------------------

typedef __attribute__((ext_vector_type(2))) float v2f;
typedef __attribute__((ext_vector_type(8))) float v8f;

#define H 4096
#define MAXLEN 15

// workspace layout (float offsets)
#define WS_LOGITS   0          // 15
#define WS_COMBINED 1024       // 8192
#define WS_X        10240      // 4096
#define WS_GI       16384      // 12288
#define WS_GH       28672      // 12288
#define WS_HNEW     40960      // 4096
#define WS_OLOG     45056      // 4096
#define WS_RED      49152      // 2

// ---------------------------------------------------------------------------
// K1: att_logits[j] = dot(att_W[j], [emb, h0]) + att_b[j]      (15 blocks)
// ---------------------------------------------------------------------------
__global__ __launch_bounds__(256) void att_logits_kernel(
    const float* __restrict__ att_W, const float* __restrict__ att_b,
    const float* __restrict__ emb_W, const int* __restrict__ ids,
    const float* __restrict__ hidden, float* __restrict__ logits)
{
    __shared__ float red[256];
    const int j = blockIdx.x;
    const int tid = threadIdx.x;
    const float* emb = emb_W + (size_t)ids[0] * H;
    const float* wr  = att_W + (size_t)j * (2 * H);
    float s = 0.0f;
    for (int i = tid; i < 2 * H; i += 256) {
        float v = (i < H) ? emb[i] : hidden[i - H];
        s += wr[i] * v;
    }
    red[tid] = s;
    __syncthreads();
    for (int off = 128; off > 0; off >>= 1) {
        if (tid < off) red[tid] += red[tid + off];
        __syncthreads();
    }
    if (tid == 0) logits[j] = red[0] + att_b[j];
}

// ---------------------------------------------------------------------------
// K2: softmax(15), att_applied = w @ e_output_seq, combined = [emb, att]
// ---------------------------------------------------------------------------
__global__ __launch_bounds__(256) void combine_kernel(
    const float* __restrict__ logits, const float* __restrict__ e_seq,
    const float* __restrict__ emb_W, const int* __restrict__ ids,
    float* __restrict__ combined, float* __restrict__ att_w_out)
{
    float l[MAXLEN];
    float m = -1e30f;
#pragma unroll
    for (int j = 0; j < MAXLEN; ++j) { l[j] = logits[j]; m = fmaxf(m, l[j]); }
    float s = 0.0f;
#pragma unroll
    for (int j = 0; j < MAXLEN; ++j) { l[j] = expf(l[j] - m); s += l[j]; }
    const float inv = 1.0f / s;

    const int i = blockIdx.x * 256 + threadIdx.x;
    if (blockIdx.x == 0 && threadIdx.x < MAXLEN)
        att_w_out[threadIdx.x] = l[threadIdx.x] * inv;

    const float* emb = emb_W + (size_t)ids[0] * H;
    combined[i] = emb[i];
    float a = 0.0f;
#pragma unroll
    for (int j = 0; j < MAXLEN; ++j) a += l[j] * e_seq[(size_t)j * H + i];
    combined[H + i] = a * inv;
}

// ---------------------------------------------------------------------------
// WMMA fp32 GEMV: out[r] = act(dot(W[r,:K], vec) + bias[r]) for 16 rows/block.
// block = 256 threads = 8 waves; wave w covers K/8 of the reduction (bounds
// scalarized via readfirstlane so the loop is SALU-controlled and unrolls).
// A-tile (16x4 fp32, ISA layout): lane<16 -> W[R0+lane][k0..k0+1],
//                                 lane>=16 -> W[R0+lane-16][k0+2..k0+3].
// B-tile broadcast: lane<16 -> (x[k0],x[k0+1]); lane>=16 -> (x[k0+2],x[k0+3]).
// All 16 D columns equal the dot partials; lane0/lane16 hold rows 0..7/8..15.
// ---------------------------------------------------------------------------
template<int KVAL, int ACT>
__global__ __launch_bounds__(256) void wmma_gemv16(
    const float* __restrict__ W, const float* __restrict__ vec,
    const float* __restrict__ bias, float* __restrict__ out)
{
    __shared__ __align__(16) float xs[KVAL];
    __shared__ float red[8][16];

    const int tid  = threadIdx.x;
    const int lane = tid & 31;

    // stage vec into LDS, 16B vectors
#pragma unroll
    for (int i = 0; i < KVAL / 4 / 256; ++i)
        ((float4*)xs)[tid + i * 256] = ((const float4*)vec)[tid + i * 256];
    __syncthreads();

    // wave-uniform K partition (scalar)
    const int wv   = __builtin_amdgcn_readfirstlane(tid >> 5);
    const int kbeg = wv * (KVAL / 8);

    const int r    = (blockIdx.x << 4) + (lane & 15);
    const int coff = (lane >> 4) << 1;            // 0 or 2
    const float* wrow = W + (size_t)r * (size_t)KVAL + coff + kbeg;
    const float* xrow = xs + coff + kbeg;

    v8f acc0 = {};
    v8f acc1 = {};
#pragma unroll 1
    for (int k0 = 0; k0 < KVAL / 8; k0 += 64) {
        __builtin_prefetch(wrow + k0 + 1024, 0, 0);   // global_prefetch_b8 (spec)
#pragma unroll
        for (int kk = 0; kk < 64; kk += 8) {
            v2f a0 = *(const v2f*)(wrow + k0 + kk);       // global b64 stream
            v2f a1 = *(const v2f*)(wrow + k0 + kk + 4);
            v2f b0 = *(const v2f*)(xrow + k0 + kk);       // ds b64 broadcast
            v2f b1 = *(const v2f*)(xrow + k0 + kk + 4);
            acc0 = __builtin_amdgcn_wmma_f32_16x16x4_f32(
                false, a0, false, b0, (short)0, acc0, false, false);
            acc1 = __builtin_amdgcn_wmma_f32_16x16x4_f32(
                false, a1, false, b1, (short)0, acc1, false, false);
        }
    }
    v8f acc = acc0 + acc1;

    if (lane == 0) {
#pragma unroll
        for (int v = 0; v < 8; ++v) red[wv][v] = acc[v];
    }
    if (lane == 16) {
#pragma unroll
        for (int v = 0; v < 8; ++v) red[wv][8 + v] = acc[v];
    }
    __syncthreads();

    if (tid < 16) {
        float s = 0.0f;
#pragma unroll
        for (int w = 0; w < 8; ++w) s += red[w][tid];
        const int row = (blockIdx.x << 4) + tid;
        s += bias[row];
        if (ACT == 1) s = fmaxf(s, 0.0f);
        out[row] = s;
    }
}

// ---------------------------------------------------------------------------
// K5: GRU gates (PyTorch r,z,n order)
// ---------------------------------------------------------------------------
__global__ __launch_bounds__(256) void gru_gate_kernel(
    const float* __restrict__ gi, const float* __restrict__ gh,
    const float* __restrict__ hidden,
    float* __restrict__ h_out_d, float* __restrict__ h_out_ws)
{
    const int i = blockIdx.x * 256 + threadIdx.x;
    const float r = 1.0f / (1.0f + expf(-(gi[i] + gh[i])));
    const float z = 1.0f / (1.0f + expf(-(gi[H + i] + gh[H + i])));
    const float n = tanhf(gi[2 * H + i] + r * gh[2 * H + i]);
    const float h = (1.0f - z) * n + z * hidden[i];
    h_out_d[i]  = h;
    h_out_ws[i] = h;
}

// ---------------------------------------------------------------------------
// K7: max + sum(exp) over 4096 logits (single block)
// ---------------------------------------------------------------------------
__global__ __launch_bounds__(1024) void lse_reduce_kernel(
    const float* __restrict__ lg, float* __restrict__ red2)
{
    __shared__ float red[1024];
    const int tid = threadIdx.x;
    float m = -1e30f;
    for (int i = tid; i < H; i += 1024) m = fmaxf(m, lg[i]);
    red[tid] = m;
    __syncthreads();
    for (int off = 512; off > 0; off >>= 1) {
        if (tid < off) red[tid] = fmaxf(red[tid], red[tid + off]);
        __syncthreads();
    }
    m = red[0];
    __syncthreads();
    float s = 0.0f;
    for (int i = tid; i < H; i += 1024) s += expf(lg[i] - m);
    red[tid] = s;
    __syncthreads();
    for (int off = 512; off > 0; off >>= 1) {
        if (tid < off) red[tid] += red[tid + off];
        __syncthreads();
    }
    if (tid == 0) { red2[0] = m; red2[1] = red[0]; }
}

// ---------------------------------------------------------------------------
// K8: out[i] = logit[i] - max - log(sum)
// ---------------------------------------------------------------------------
__global__ __launch_bounds__(256) void lsm_final_kernel(
    const float* __restrict__ lg, const float* __restrict__ red2,
    float* __restrict__ out)
{
    const int i = blockIdx.x * 256 + threadIdx.x;
    out[i] = lg[i] - red2[0] - logf(red2[1]);
}

// ---------------------------------------------------------------------------
extern "C" void kernel_launch(void* const* d_in, const int* in_sizes, int n_in,
                              void* d_out, int out_size, void* d_ws, size_t ws_size,
                              hipStream_t stream)
{
    const int*   ids    = (const int*)  d_in[0];
    const float* hidden = (const float*)d_in[1];
    // d_in[2] e_output: unused by reference
    const float* e_seq  = (const float*)d_in[3];
    const float* emb_W  = (const float*)d_in[4];
    const float* att_W  = (const float*)d_in[5];
    const float* att_b  = (const float*)d_in[6];
    const float* comb_W = (const float*)d_in[7];
    const float* comb_b = (const float*)d_in[8];
    const float* Wih    = (const float*)d_in[9];
    const float* Whh    = (const float*)d_in[10];
    const float* bih    = (const float*)d_in[11];
    const float* bhh    = (const float*)d_in[12];
    const float* outW   = (const float*)d_in[13];
    const float* outb   = (const float*)d_in[14];

    float* out = (float*)d_out;   // [0,4096) log_softmax; [4096,8192) h_new; [8192,8207) att_w
    float* ws  = (float*)d_ws;

    att_logits_kernel<<<MAXLEN, 256, 0, stream>>>(att_W, att_b, emb_W, ids, hidden,
                                                  ws + WS_LOGITS);
    combine_kernel<<<H / 256, 256, 0, stream>>>(ws + WS_LOGITS, e_seq, emb_W, ids,
                                                ws + WS_COMBINED, out + 2 * H);
    // x = relu(comb_W @ combined + comb_b)   [4096 x 8192]
    wmma_gemv16<2 * H, 1><<<H / 16, 256, 0, stream>>>(comb_W, ws + WS_COMBINED,
                                                      comb_b, ws + WS_X);
    // gi = gru_W_ih @ x,  gh = gru_W_hh @ h0   [12288 x 4096] each
    wmma_gemv16<H, 0><<<(3 * H) / 16, 256, 0, stream>>>(Wih, ws + WS_X, bih,
                                                        ws + WS_GI);
    wmma_gemv16<H, 0><<<(3 * H) / 16, 256, 0, stream>>>(Whh, hidden, bhh,
                                                        ws + WS_GH);
    gru_gate_kernel<<<H / 256, 256, 0, stream>>>(ws + WS_GI, ws + WS_GH, hidden,
                                                 out + H, ws + WS_HNEW);
    // logits = out_W @ h_new + out_b   [4096 x 4096]
    wmma_gemv16<H, 0><<<H / 16, 256, 0, stream>>>(outW, ws + WS_HNEW, outb,
                                                  ws + WS_OLOG);
    lse_reduce_kernel<<<1, 1024, 0, stream>>>(ws + WS_OLOG, ws + WS_RED);
    lsm_final_kernel<<<H / 256, 256, 0, stream>>>(ws + WS_OLOG, ws + WS_RED, out);
}